// GNN_20392504721510
// MI455X (gfx1250) — compile-verified
//
#include <hip/hip_runtime.h>

typedef __attribute__((ext_vector_type(2))) float v2f;
typedef __attribute__((ext_vector_type(8))) float v8f;

#define IN_DIM 512
#define HID    16
#define ODIM   64

// ---------- degree / normalization ----------
__global__ void k_deg_init(float* __restrict__ deg, int n) {
  int i = blockIdx.x * blockDim.x + threadIdx.x;
  if (i < n) deg[i] = 1.0f;  // self-loop contributes 1 to every dst degree
}

__global__ void k_deg_edges(const int* __restrict__ dst, float* __restrict__ deg, int E) {
  int i = blockIdx.x * blockDim.x + threadIdx.x;
  if (i < E) atomicAdd(&deg[dst[i]], 1.0f);
}

__global__ void k_dinv(float* __restrict__ deg, int n) {
  int i = blockIdx.x * blockDim.x + threadIdx.x;
  if (i < n) {
    float d = deg[i];
    deg[i] = (d > 0.0f) ? (1.0f / sqrtf(d)) : 0.0f;
  }
}

// ---------- GEMM1: h1[N,16] = x[N,512] @ W1[512,16] via V_WMMA_F32_16X16X4_F32 ----------
__global__ void k_gemm1(const float* __restrict__ x, const float* __restrict__ W1,
                        float* __restrict__ h1, int n) {
  __shared__ float lw[IN_DIM * HID];  // 32 KB
  for (int i = threadIdx.x; i < IN_DIM * HID; i += blockDim.x) lw[i] = W1[i];
  __syncthreads();

  int wave  = (blockIdx.x * blockDim.x + threadIdx.x) >> 5;
  int lane  = threadIdx.x & 31;
  int tiles = (n + 15) >> 4;
  if (wave >= tiles) return;

  int half = lane >> 4;   // 0: K pair {0,1}; 1: K pair {2,3}
  int ml   = lane & 15;   // A row / B column within tile
  int row  = wave * 16 + ml;
  int rrow = (row < n) ? row : (n - 1);
  const float* xr = x + (size_t)rrow * IN_DIM;

  v8f acc = {0.f, 0.f, 0.f, 0.f, 0.f, 0.f, 0.f, 0.f};
  for (int k = 0; k < IN_DIM; k += 4) {
    int ka = k + 2 * half;
    v2f a; a.x = xr[ka];                 a.y = xr[ka + 1];
    v2f b; b.x = lw[ka * HID + ml];      b.y = lw[(ka + 1) * HID + ml];
    acc = __builtin_amdgcn_wmma_f32_16x16x4_f32(false, a, false, b, (short)0, acc,
                                                false, false);
  }
#pragma unroll
  for (int r = 0; r < 8; ++r) {
    int orow = wave * 16 + r + 8 * half;
    if (orow < n) h1[(size_t)orow * HID + ml] = acc[r];
  }
}

// ---------- z1 init: self-loop (norm = dinv^2) + bias ----------
__global__ void k_init_z1(const float* __restrict__ h1, const float* __restrict__ dinv,
                          const float* __restrict__ b1, float* __restrict__ z1, int n) {
  int t = blockIdx.x * blockDim.x + threadIdx.x;
  int total = n * HID;
  if (t < total) {
    int i = t / HID, j = t - i * HID;
    float di = dinv[i];
    z1[t] = h1[t] * di * di + b1[j];
  }
}

// ---------- edge aggregation layer 1: thread = (edge, float4 chunk of 16 feats) ----------
__global__ void k_agg1(const int* __restrict__ src, const int* __restrict__ dst,
                       const float* __restrict__ dinv, const float* __restrict__ h1,
                       float* __restrict__ z1, int E) {
  int total  = E * 4;
  int stride = gridDim.x * blockDim.x;
  for (int t = blockIdx.x * blockDim.x + threadIdx.x; t < total; t += stride) {
    int e = t >> 2, c = t & 3;
    int s = src[e], d = dst[e];
    float norm = dinv[s] * dinv[d];
    const float4 hv = *(const float4*)(h1 + (size_t)s * HID + c * 4);
    float* o = z1 + (size_t)d * HID + c * 4;
    atomicAdd(o + 0, hv.x * norm);
    atomicAdd(o + 1, hv.y * norm);
    atomicAdd(o + 2, hv.z * norm);
    atomicAdd(o + 3, hv.w * norm);
  }
}

// ---------- GEMM2: h2[N,64] = relu(z1)[N,16] @ W2[16,64] via WMMA ----------
__global__ void k_gemm2(const float* __restrict__ z1, const float* __restrict__ W2,
                        float* __restrict__ h2, int n) {
  __shared__ float lw[HID * ODIM];  // 4 KB
  for (int i = threadIdx.x; i < HID * ODIM; i += blockDim.x) lw[i] = W2[i];
  __syncthreads();

  int wave  = (blockIdx.x * blockDim.x + threadIdx.x) >> 5;
  int lane  = threadIdx.x & 31;
  int tiles = (n + 15) >> 4;
  if (wave >= tiles) return;

  int half = lane >> 4;
  int ml   = lane & 15;
  int row  = wave * 16 + ml;
  int rrow = (row < n) ? row : (n - 1);
  const float* zr = z1 + (size_t)rrow * HID;

  v2f afrag[4];
#pragma unroll
  for (int kk = 0; kk < 4; ++kk) {
    int ka = kk * 4 + 2 * half;
    afrag[kk].x = fmaxf(zr[ka], 0.f);       // fused ReLU
    afrag[kk].y = fmaxf(zr[ka + 1], 0.f);
  }

#pragma unroll
  for (int tcol = 0; tcol < 4; ++tcol) {    // N = 64 -> 4 column tiles
    v8f acc = {0.f, 0.f, 0.f, 0.f, 0.f, 0.f, 0.f, 0.f};
#pragma unroll
    for (int kk = 0; kk < 4; ++kk) {
      int ka = kk * 4 + 2 * half;
      v2f b;
      b.x = lw[ka * ODIM + tcol * 16 + ml];
      b.y = lw[(ka + 1) * ODIM + tcol * 16 + ml];
      acc = __builtin_amdgcn_wmma_f32_16x16x4_f32(false, afrag[kk], false, b, (short)0,
                                                  acc, false, false);
    }
#pragma unroll
    for (int r = 0; r < 8; ++r) {
      int orow = wave * 16 + r + 8 * half;
      if (orow < n) h2[(size_t)orow * ODIM + tcol * 16 + ml] = acc[r];
    }
  }
}

// ---------- out init: self-loop + bias ----------
__global__ void k_init_out(const float* __restrict__ h2, const float* __restrict__ dinv,
                           const float* __restrict__ b2, float* __restrict__ out, int n) {
  int t = blockIdx.x * blockDim.x + threadIdx.x;
  int total = n * ODIM;
  if (t < total) {
    int i = t / ODIM, j = t - i * ODIM;
    float di = dinv[i];
    out[t] = h2[t] * di * di + b2[j];
  }
}

// ---------- edge aggregation layer 2: thread = (edge, float4 chunk of 64 feats) ----------
__global__ void k_agg2(const int* __restrict__ src, const int* __restrict__ dst,
                       const float* __restrict__ dinv, const float* __restrict__ h2,
                       float* __restrict__ out, int E) {
  int total  = E * 16;
  int stride = gridDim.x * blockDim.x;
  for (int t = blockIdx.x * blockDim.x + threadIdx.x; t < total; t += stride) {
    int e = t >> 4, c = t & 15;
    int s = src[e], d = dst[e];
    float norm = dinv[s] * dinv[d];
    const float4 hv = *(const float4*)(h2 + (size_t)s * ODIM + c * 4);
    float* o = out + (size_t)d * ODIM + c * 4;
    atomicAdd(o + 0, hv.x * norm);
    atomicAdd(o + 1, hv.y * norm);
    atomicAdd(o + 2, hv.z * norm);
    atomicAdd(o + 3, hv.w * norm);
  }
}

extern "C" void kernel_launch(void* const* d_in, const int* in_sizes, int n_in,
                              void* d_out, int out_size, void* d_ws, size_t ws_size,
                              hipStream_t stream) {
  const float* x  = (const float*)d_in[0];
  const float* W1 = (const float*)d_in[1];
  const float* b1 = (const float*)d_in[2];
  const float* W2 = (const float*)d_in[3];
  const float* b2 = (const float*)d_in[4];
  const int*   ei = (const int*)d_in[5];

  const int N = in_sizes[0] / IN_DIM;
  const int E = in_sizes[5] / 2;
  const int* src = ei;        // edge_index[0]
  const int* dst = ei + E;    // edge_index[1]
  float* out = (float*)d_out;

  char* ws = (char*)d_ws;
  size_t off = 0;
  auto alloc = [&](size_t nbytes) -> void* {
    void* p = ws + off;
    off = (off + nbytes + 255) & ~(size_t)255;
    return p;
  };
  float* dinv = (float*)alloc((size_t)N * sizeof(float));
  float* h1   = (float*)alloc((size_t)N * HID  * sizeof(float));
  float* z1   = (float*)alloc((size_t)N * HID  * sizeof(float));
  float* h2   = (float*)alloc((size_t)N * ODIM * sizeof(float));

  const int B = 256;                       // 8 wave32s per block
  const int tiles = (N + 15) >> 4;
  const int gemmBlocks = (tiles + 7) / 8;  // one wave per 16-row tile

  k_deg_init <<<(N + B - 1) / B, B, 0, stream>>>(dinv, N);
  k_deg_edges<<<(E + B - 1) / B, B, 0, stream>>>(dst, dinv, E);
  k_dinv     <<<(N + B - 1) / B, B, 0, stream>>>(dinv, N);

  k_gemm1    <<<gemmBlocks, B, 0, stream>>>(x, W1, h1, N);
  k_init_z1  <<<((size_t)N * HID + B - 1) / B, B, 0, stream>>>(h1, dinv, b1, z1, N);
  k_agg1     <<<(E * 4 + B - 1) / B, B, 0, stream>>>(src, dst, dinv, h1, z1, E);

  k_gemm2    <<<gemmBlocks, B, 0, stream>>>(z1, W2, h2, N);
  k_init_out <<<((size_t)N * ODIM + B - 1) / B, B, 0, stream>>>(h2, dinv, b2, out, N);
  k_agg2     <<<(E * 16 + B - 1) / B, B, 0, stream>>>(src, dst, dinv, h2, out, E);
}